// AdaptiveRepVGGDW_79697413145226
// MI455X (gfx1250) — compile-verified
//
#include <hip/hip_runtime.h>
#include <math.h>

typedef __attribute__((ext_vector_type(2))) float v2f;
typedef __attribute__((ext_vector_type(8))) float v8f;
typedef __attribute__((__vector_size__(16))) int v4i_raw;

#define B_    64
#define C_    256
#define HW_   1024
#define NHW_  65536.0f
#define EPS_  1e-5f

// tile geometry: 36 rows x 40-float stride; interior (h,w) at (h+2)*40 + 4 + w
#define TSTRIDE 40
#define TROWS   36
#define TSIZE   (TSTRIDE * TROWS)

// workspace layout (float offsets)
enum {
  WS_POOLED = 0,            // [B*C] = 16384
  WS_SUM3   = 16384,
  WS_SQ3    = 16384 + 256,
  WS_SUM5   = 16384 + 512,
  WS_SQ5    = 16384 + 768,
  WS_SUMO   = 16384 + 1024,
  WS_SQO    = 16384 + 1280,
  WS_S3     = 16384 + 1536,
  WS_T3     = 16384 + 1792,
  WS_S5     = 16384 + 2048,
  WS_T5     = 16384 + 2304,
  WS_SF     = 16384 + 2560,
  WS_TF     = 16384 + 2816,
  WS_KW     = 16384 + 3072  // [B*2]
};

__device__ __forceinline__ float wave_reduce(float v) {
  #pragma unroll
  for (int o = 16; o > 0; o >>= 1) v += __shfl_down(v, o, 32);
  return v;
}

// valid on tid 0 only; 256-thread blocks (8 waves)
__device__ __forceinline__ float block_reduce_sum(float v, float* scratch) {
  int tid = threadIdx.x;
  float w = wave_reduce(v);
  __syncthreads();                       // protect scratch from previous call
  if ((tid & 31) == 0) scratch[tid >> 5] = w;
  __syncthreads();
  float r = 0.f;
  if (tid < 32) {
    float t = (tid < 8) ? scratch[tid] : 0.f;
    r = wave_reduce(t);
  }
  return r;
}

// cooperative zero-padded tile fill; 256 threads, one 16B chunk each.
// Uses CDNA5 async global->LDS (ASYNCcnt path) when available, else
// aligned float4 load + ds_store_b128.
__device__ __forceinline__ void load_tile(const float* __restrict__ plane,
                                          float* tile) {
  const int tid = threadIdx.x;
  for (int i = tid; i < TSIZE; i += 256) tile[i] = 0.f;
  __syncthreads();                      // commit zeros (ds) before async writes
  const int h  = tid >> 3;              // 0..31
  const int w4 = tid & 7;               // 0..7 (float4 column)
  float* ldst = &tile[(h + 2) * TSTRIDE + 4 + w4 * 4];   // 16B-aligned
#if defined(__gfx1250__) && __has_builtin(__builtin_amdgcn_global_load_async_to_lds_b128)
  typedef __attribute__((address_space(1))) v4i_raw* gp128_t;
  typedef __attribute__((address_space(3))) v4i_raw* lp128_t;
  __builtin_amdgcn_global_load_async_to_lds_b128(
      (gp128_t)(void*)(plane + tid * 4),
      (lp128_t)(void*)ldst,
      /*offset=*/0, /*cpol=*/0);
  #if __has_builtin(__builtin_amdgcn_s_wait_asynccnt)
  __builtin_amdgcn_s_wait_asynccnt(0);
  #else
  asm volatile("s_wait_asynccnt 0x0" ::: "memory");
  #endif
#else
  const float4 v = ((const float4*)plane)[tid];
  *(float4*)ldst = v;
#endif
  __syncthreads();
}

// ---------------- K0: zero accumulators ----------------
__global__ void k_zero(float* ws) {
  int i = threadIdx.x;
  for (int o = i; o < 6 * C_; o += 256) ws[WS_SUM3 + o] = 0.f;
}

// ---------------- K1: stencil stats pass ----------------
__global__ __launch_bounds__(256) void k_stats(const float* __restrict__ x,
                                               const float* __restrict__ w3,
                                               const float* __restrict__ w5,
                                               float* __restrict__ ws) {
  __shared__ float tile[TSIZE];
  __shared__ float sw3[9], sw5[25];
  __shared__ float scratch[8];
  const int blk = blockIdx.x;           // b*C + c
  const int c   = blk & (C_ - 1);
  const int tid = threadIdx.x;
  const float* plane = x + ((size_t)blk << 10);

  if (tid < 9)  sw3[tid] = w3[c * 9 + tid];
  if (tid < 25) sw5[tid] = w5[c * 25 + tid];
  load_tile(plane, tile);

  float s3 = 0.f, q3 = 0.f, s5 = 0.f, q5 = 0.f, sx = 0.f;
  for (int p = tid; p < HW_; p += 256) {
    int h = p >> 5, w = p & 31;
    const float* t = &tile[(h + 2) * TSTRIDE + (w + 4)];
    float c3 = 0.f;
    #pragma unroll
    for (int dy = -1; dy <= 1; ++dy)
      #pragma unroll
      for (int dx = -1; dx <= 1; ++dx)
        c3 = fmaf(sw3[(dy + 1) * 3 + (dx + 1)], t[dy * TSTRIDE + dx], c3);
    float c5 = 0.f;
    #pragma unroll
    for (int dy = -2; dy <= 2; ++dy)
      #pragma unroll
      for (int dx = -2; dx <= 2; ++dx)
        c5 = fmaf(sw5[(dy + 2) * 5 + (dx + 2)], t[dy * TSTRIDE + dx], c5);
    s3 += c3; q3 += c3 * c3; s5 += c5; q5 += c5 * c5; sx += t[0];
  }
  float r;
  r = block_reduce_sum(s3, scratch); if (tid == 0) atomicAdd(&ws[WS_SUM3 + c], r);
  r = block_reduce_sum(q3, scratch); if (tid == 0) atomicAdd(&ws[WS_SQ3  + c], r);
  r = block_reduce_sum(s5, scratch); if (tid == 0) atomicAdd(&ws[WS_SUM5 + c], r);
  r = block_reduce_sum(q5, scratch); if (tid == 0) atomicAdd(&ws[WS_SQ5  + c], r);
  r = block_reduce_sum(sx, scratch); if (tid == 0) ws[WS_POOLED + blk] = r * (1.f / HW_);
}

// ---------------- K2: BN finalize + kernel predictor (WMMA) ----------------
// 1 block, 128 threads = 4 waves
__global__ __launch_bounds__(128) void k_predict(float* __restrict__ ws,
                                                 const float* __restrict__ bn3_g, const float* __restrict__ bn3_b,
                                                 const float* __restrict__ bn5_g, const float* __restrict__ bn5_b,
                                                 const float* __restrict__ kp1_w,
                                                 const float* __restrict__ kpbn1_g, const float* __restrict__ kpbn1_b,
                                                 const float* __restrict__ kp2_w,
                                                 const float* __restrict__ kpbn2_g, const float* __restrict__ kpbn2_b) {
  // B panel staged in LDS, zero-padded to N=16, K-pairs interleaved:
  // bpad[(k>>1)*32 + n*2 + (k&1)] = (n<4) ? kp1_w[n*256+k] : 0
  __shared__ float bpad[C_ * 16];
  __shared__ float hbuf[B_ * 4];
  __shared__ float lbuf[B_ * 2];
  __shared__ float sc1[4], sh1[4], sc2[2], sh2[2];
  const int tid = threadIdx.x;

  // finalize bn3/bn5 scale+shift
  for (int c = tid; c < C_; c += 128) {
    float m3 = ws[WS_SUM3 + c] * (1.f / NHW_);
    float v3 = ws[WS_SQ3 + c] * (1.f / NHW_) - m3 * m3;
    float s3 = bn3_g[c] * rsqrtf(v3 + EPS_);
    ws[WS_S3 + c] = s3;
    ws[WS_T3 + c] = bn3_b[c] - m3 * s3;
    float m5 = ws[WS_SUM5 + c] * (1.f / NHW_);
    float v5 = ws[WS_SQ5 + c] * (1.f / NHW_) - m5 * m5;
    float s5 = bn5_g[c] * rsqrtf(v5 + EPS_);
    ws[WS_S5 + c] = s5;
    ws[WS_T5 + c] = bn5_b[c] - m5 * s5;
  }

  // stage padded B panel
  for (int i = tid; i < C_ * 16; i += 128) {
    int k = i >> 4, n = i & 15;
    bpad[(k >> 1) * 32 + n * 2 + (k & 1)] = (n < 4) ? kp1_w[n * C_ + k] : 0.f;
  }
  __syncthreads();

  // h = pooled[64,256] @ kp1_w.T[256,4]  via V_WMMA_F32_16X16X4_F32
  {
    const float* pooled = ws + WS_POOLED;
    const int wave = tid >> 5;          // 0..3 -> M tile
    const int lane = tid & 31;
    const int half = lane >> 4;
    const int mrow = lane & 15;
    const int n    = lane & 15;
    const int m    = wave * 16 + mrow;
    const float2* arow = (const float2*)(pooled + m * C_);
    v8f acc = {};
    #pragma unroll 4
    for (int k0 = 0; k0 < C_; k0 += 4) {
      int kp = (k0 >> 1) + half;        // float2 index: ka = k0 + 2*half
      float2 av = arow[kp];
      const float2 bv = *(const float2*)&bpad[kp * 32 + n * 2];
      v2f a;  a.x = av.x;  a.y = av.y;
      v2f bf; bf.x = bv.x; bf.y = bv.y;
      acc = __builtin_amdgcn_wmma_f32_16x16x4_f32(false, a, false, bf,
                                                  (short)0, acc, false, false);
    }
    if (n < 4) {
      #pragma unroll
      for (int r = 0; r < 8; ++r) {
        int mm = wave * 16 + r + 8 * half;   // D layout: VGPR r -> M = r + 8*half
        hbuf[mm * 4 + n] = acc[r];
      }
    }
  }
  __syncthreads();

  // bn1d over batch (4 cols) -> GELU(exact)
  if (tid < 4) {
    float s = 0.f, q = 0.f;
    for (int t = 0; t < B_; ++t) { float v = hbuf[t * 4 + tid]; s += v; q += v * v; }
    float mean = s * (1.f / B_);
    float var  = q * (1.f / B_) - mean * mean;
    float sc = kpbn1_g[tid] * rsqrtf(var + EPS_);
    sc1[tid] = sc;
    sh1[tid] = kpbn1_b[tid] - mean * sc;
  }
  __syncthreads();
  if (tid < B_) {
    #pragma unroll
    for (int j = 0; j < 4; ++j) {
      float v = hbuf[tid * 4 + j] * sc1[j] + sh1[j];
      hbuf[tid * 4 + j] = 0.5f * v * (1.0f + erff(v * 0.70710678118654752f));
    }
    #pragma unroll
    for (int i = 0; i < 2; ++i) {
      float l = 0.f;
      #pragma unroll
      for (int j = 0; j < 4; ++j) l = fmaf(hbuf[tid * 4 + j], kp2_w[i * 4 + j], l);
      lbuf[tid * 2 + i] = l;
    }
  }
  __syncthreads();
  if (tid < 2) {
    float s = 0.f, q = 0.f;
    for (int t = 0; t < B_; ++t) { float v = lbuf[t * 2 + tid]; s += v; q += v * v; }
    float mean = s * (1.f / B_);
    float var  = q * (1.f / B_) - mean * mean;
    float sc = kpbn2_g[tid] * rsqrtf(var + EPS_);
    sc2[tid] = sc;
    sh2[tid] = kpbn2_b[tid] - mean * sc;
  }
  __syncthreads();
  if (tid < B_) {
    float l0 = lbuf[tid * 2 + 0] * sc2[0] + sh2[0];
    float l1 = lbuf[tid * 2 + 1] * sc2[1] + sh2[1];
    float mx = fmaxf(l0, l1);
    float e0 = __expf(l0 - mx), e1 = __expf(l1 - mx);
    float inv = 1.f / (e0 + e1);
    ws[WS_KW + tid * 2 + 0] = e0 * inv;
    ws[WS_KW + tid * 2 + 1] = e1 * inv;
  }
}

// ---------------- K3: combine pass (writes unnormalized out + out stats) ----
__global__ __launch_bounds__(256) void k_combine(const float* __restrict__ x,
                                                 const float* __restrict__ w3,
                                                 const float* __restrict__ w5,
                                                 const float* __restrict__ w1,
                                                 const float* __restrict__ b1,
                                                 float* __restrict__ out,
                                                 float* __restrict__ ws) {
  __shared__ float tile[TSIZE];
  __shared__ float sw3[9], sw5[25];
  __shared__ float scratch[8];
  const int blk = blockIdx.x;
  const int c   = blk & (C_ - 1);
  const int b   = blk >> 8;
  const int tid = threadIdx.x;
  const float* plane = x + ((size_t)blk << 10);
  float* oplane = out + ((size_t)blk << 10);

  const float kw0 = ws[WS_KW + b * 2 + 0];
  const float kw1 = ws[WS_KW + b * 2 + 1];
  const float s3v = ws[WS_S3 + c], t3v = ws[WS_T3 + c];
  const float s5v = ws[WS_S5 + c], t5v = ws[WS_T5 + c];
  const float w1c = w1[c], b1c = b1[c];

  if (tid < 9)  sw3[tid] = w3[c * 9 + tid];
  if (tid < 25) sw5[tid] = w5[c * 25 + tid];
  load_tile(plane, tile);

  float so = 0.f, qo = 0.f;
  for (int p = tid; p < HW_; p += 256) {
    int h = p >> 5, w = p & 31;
    const float* t = &tile[(h + 2) * TSTRIDE + (w + 4)];
    float c3 = 0.f;
    #pragma unroll
    for (int dy = -1; dy <= 1; ++dy)
      #pragma unroll
      for (int dx = -1; dx <= 1; ++dx)
        c3 = fmaf(sw3[(dy + 1) * 3 + (dx + 1)], t[dy * TSTRIDE + dx], c3);
    float c5 = 0.f;
    #pragma unroll
    for (int dy = -2; dy <= 2; ++dy)
      #pragma unroll
      for (int dx = -2; dx <= 2; ++dx)
        c5 = fmaf(sw5[(dy + 2) * 5 + (dx + 2)], t[dy * TSTRIDE + dx], c5);
    float o = kw0 * fmaf(c3, s3v, t3v) + kw1 * fmaf(c5, s5v, t5v)
            + fmaf(t[0], w1c, b1c);
    oplane[p] = o;
    so += o; qo += o * o;
  }
  float r;
  r = block_reduce_sum(so, scratch); if (tid == 0) atomicAdd(&ws[WS_SUMO + c], r);
  r = block_reduce_sum(qo, scratch); if (tid == 0) atomicAdd(&ws[WS_SQO  + c], r);
}

// ---------------- K4: finalize output BN scale/shift ----------------
__global__ void k_final_bn(float* __restrict__ ws,
                           const float* __restrict__ bn_g,
                           const float* __restrict__ bn_b) {
  int c = threadIdx.x;
  float m = ws[WS_SUMO + c] * (1.f / NHW_);
  float v = ws[WS_SQO + c] * (1.f / NHW_) - m * m;
  float s = bn_g[c] * rsqrtf(v + EPS_);
  ws[WS_SF + c] = s;
  ws[WS_TF + c] = bn_b[c] - m * s;
}

// ---------------- K5: in-place normalize (float4) ----------------
__global__ __launch_bounds__(256) void k_normalize(float4* __restrict__ out,
                                                   const float* __restrict__ ws) {
  size_t i4 = (size_t)blockIdx.x * blockDim.x + threadIdx.x;  // 4.19M float4s
  size_t idx = i4 << 2;
  int c = (int)((idx >> 10) & (C_ - 1));
  float s = ws[WS_SF + c], t = ws[WS_TF + c];
  float4 v = out[i4];
  v.x = fmaf(v.x, s, t); v.y = fmaf(v.y, s, t);
  v.z = fmaf(v.z, s, t); v.w = fmaf(v.w, s, t);
  out[i4] = v;
}

extern "C" void kernel_launch(void* const* d_in, const int* in_sizes, int n_in,
                              void* d_out, int out_size, void* d_ws, size_t ws_size,
                              hipStream_t stream) {
  const float* x       = (const float*)d_in[0];
  const float* conv3_w = (const float*)d_in[1];
  const float* bn3_g   = (const float*)d_in[2];
  const float* bn3_b   = (const float*)d_in[3];
  const float* conv5_w = (const float*)d_in[4];
  const float* bn5_g   = (const float*)d_in[5];
  const float* bn5_b   = (const float*)d_in[6];
  const float* conv1_w = (const float*)d_in[7];
  const float* conv1_b = (const float*)d_in[8];
  const float* bn_g    = (const float*)d_in[9];
  const float* bn_b    = (const float*)d_in[10];
  const float* kp1_w   = (const float*)d_in[11];
  const float* kpbn1_g = (const float*)d_in[12];
  const float* kpbn1_b = (const float*)d_in[13];
  const float* kp2_w   = (const float*)d_in[14];
  const float* kpbn2_g = (const float*)d_in[15];
  const float* kpbn2_b = (const float*)d_in[16];
  float* out = (float*)d_out;
  float* ws  = (float*)d_ws;

  const int nplanes = B_ * C_;  // 16384

  k_zero<<<1, 256, 0, stream>>>(ws);
  k_stats<<<nplanes, 256, 0, stream>>>(x, conv3_w, conv5_w, ws);
  k_predict<<<1, 128, 0, stream>>>(ws, bn3_g, bn3_b, bn5_g, bn5_b,
                                   kp1_w, kpbn1_g, kpbn1_b,
                                   kp2_w, kpbn2_g, kpbn2_b);
  k_combine<<<nplanes, 256, 0, stream>>>(x, conv3_w, conv5_w, conv1_w, conv1_b,
                                         out, ws);
  k_final_bn<<<1, 256, 0, stream>>>(ws, bn_g, bn_b);
  k_normalize<<<(B_ * C_ * HW_ / 4) / 256, 256, 0, stream>>>((float4*)out, ws);
}